// CharEncoder_61624190763800
// MI455X (gfx1250) — compile-verified
//
#include <hip/hip_runtime.h>

typedef __attribute__((ext_vector_type(2))) float v2f;
typedef __attribute__((ext_vector_type(8))) float v8f;

#define SS 512
#define WW 64
#define CC 16
#define VOCAB 258
#define EE 64
#define HH 8
#define NG 32              // 4*H gates
#define NWORD (SS * WW)    // 32768
#define MT 17              // ceil(258/16) row tiles
#define GROWS (MT * 16)    // 272 padded vocab rows in ws
#define LSTRIDE 36         // LDS row stride (floats): 144B, 16B-aligned, bank-spreading

__device__ __forceinline__ float fsig(float x) {
    return __builtin_amdgcn_rcpf(1.0f + __builtin_amdgcn_exp2f(-1.44269504f * x));
}
__device__ __forceinline__ float ftanhf(float x) {
    float e = __builtin_amdgcn_exp2f(2.88539008f * x);  // exp(2x)
    return (e - 1.0f) * __builtin_amdgcn_rcpf(e + 1.0f);
}

// ---------------------------------------------------------------------------
// Kernel A: per-vocab gate tables  G[d][272][32] = emb_z @ W_ih^T + (b_ih+b_hh)
// One wave per 16x16 output tile, V_WMMA_F32_16X16X4_F32, K=64 in 16 steps.
// ---------------------------------------------------------------------------
__global__ __launch_bounds__(32) void build_tables_kernel(
    const float* __restrict__ emb,
    const float* __restrict__ Wf, const float* __restrict__ bif, const float* __restrict__ bhf,
    const float* __restrict__ Wb, const float* __restrict__ bib, const float* __restrict__ bhb,
    float* __restrict__ G)
{
    const int tile = blockIdx.x;        // 0..67
    const int d    = tile / (MT * 2);   // direction
    const int r    = tile % (MT * 2);
    const int mt   = r >> 1;            // vocab row tile
    const int nt   = r & 1;             // gate col tile

    const float* Wih = d ? Wb : Wf;
    const float* bi  = d ? bib : bif;
    const float* bh  = d ? bhb : bhf;

    const int lane = threadIdx.x & 31;
    const int half = lane >> 4;         // 0: K pair {k0,k0+1}, 1: {k0+2,k0+3}
    const int l16  = lane & 15;

    const int row = mt * 16 + l16;                   // emb row for A (M index)
    const int col = nt * 16 + l16;                   // gate row for B (N index)
    const int rc  = row < VOCAB ? row : (VOCAB - 1); // clamp to avoid OOB loads
    const bool valid = (row > 0) && (row < VOCAB);   // padding_idx=0 -> zero row

    v8f acc = {};
    #pragma unroll
    for (int k0 = 0; k0 < EE; k0 += 4) {
        const int ka = k0 + 2 * half;
        float a0 = emb[rc * EE + ka + 0];
        float a1 = emb[rc * EE + ka + 1];
        a0 = valid ? a0 : 0.0f;
        a1 = valid ? a1 : 0.0f;
        float b0 = Wih[col * EE + ka + 0];
        float b1 = Wih[col * EE + ka + 1];
        v2f av = {a0, a1};
        v2f bv = {b0, b1};
        acc = __builtin_amdgcn_wmma_f32_16x16x4_f32(
            false, av, false, bv, (short)0, acc, false, false);
    }

    const float bias = bi[col] + bh[col];
    #pragma unroll
    for (int j = 0; j < 8; ++j) {
        const int m = mt * 16 + j + 8 * half;
        G[(d * GROWS + m) * NG + (nt * 16 + l16)] = acc[j] + bias;
    }
}

// ---------------------------------------------------------------------------
// Kernel B: one (word, direction) per lane; direction is block-uniform.
// Gate table in LDS via async global->LDS B128 copies; W_hh pair-interleaved
// for packed (v_pk_fma_f32) accumulation; h/c/acc in registers; fixed 16-step
// fully-unrolled recurrence with s<len masking.
// ---------------------------------------------------------------------------
template <bool REV>
__device__ __forceinline__ void run_dir(const float* __restrict__ sG,
                                        const float* __restrict__ sW2,
                                        const int* ch, int len, float* acc)
{
    float h[HH], c[HH];
    #pragma unroll
    for (int k = 0; k < HH; ++k) { h[k] = 0.0f; c[k] = 0.0f; acc[k] = 0.0f; }

    #pragma unroll
    for (int step = 0; step < CC; ++step) {
        const int  s   = REV ? (CC - 1 - step) : step;
        const bool act = (s < len);
        const float* gr = sG + ch[s] * LSTRIDE;

        v2f g2[16];
        #pragma unroll
        for (int q = 0; q < 8; ++q) {
            float4 v = *(const float4*)(gr + 4 * q);   // random-row LDS gather
            v2f lo = {v.x, v.y};
            v2f hi = {v.z, v.w};
            g2[2 * q]     = lo;
            g2[2 * q + 1] = hi;
        }
        // g += h @ W_hh^T  as packed pairs over the gate dimension
        #pragma unroll
        for (int k = 0; k < HH; ++k) {
            v2f hk = {h[k], h[k]};
            #pragma unroll
            for (int j = 0; j < 16; ++j) {
                v2f w = *(const v2f*)(sW2 + ((k << 4) + j) * 2);
                g2[j] += hk * w;
            }
        }
        #pragma unroll
        for (int k = 0; k < HH; ++k) {
            const float ig = fsig(g2[k >> 1][k & 1]);              // gates 0..7
            const float fg = fsig(g2[(HH + k) >> 1][k & 1]);       // gates 8..15
            const float gg = ftanhf(g2[(2 * HH + k) >> 1][k & 1]); // gates 16..23
            const float og = fsig(g2[(3 * HH + k) >> 1][k & 1]);   // gates 24..31
            const float nc = fg * c[k] + ig * gg;
            const float nh = og * ftanhf(nc);
            c[k] = act ? nc : c[k];
            h[k] = act ? nh : h[k];
            acc[k] += act ? nh : 0.0f;
        }
    }
}

__global__ __launch_bounds__(256) void encode_kernel(
    const int* __restrict__ chars, const int* __restrict__ counts,
    const float* __restrict__ G,
    const float* __restrict__ Whf, const float* __restrict__ Whb,
    float* __restrict__ out)
{
    __shared__ float sG[VOCAB * LSTRIDE];   // 37,152 B (one direction)
    __shared__ float sW2[NG * HH];          //  1,024 B pair-interleaved W_hh

    const int tid  = threadIdx.x;
    const int dir  = blockIdx.x >> 7;                      // block-uniform
    const int widx = (blockIdx.x & 127) * 256 + tid;       // word index

    __builtin_prefetch(chars + widx * CC, 0, 0);           // global_prefetch_b8

    const float* gsrc = G + (size_t)dir * GROWS * NG;
    const float* wsrc = dir ? Whb : Whf;

    // ---- async global->LDS table copy (B128, both sides 16B aligned) ----
    for (int q = tid; q < VOCAB * 8; q += 256) {
        const int rr = q >> 3, qc = q & 7;
        unsigned lds   = (unsigned)(unsigned long long)(sG + rr * LSTRIDE + qc * 4);
        int      goff  = (rr * NG + qc * 4) * 4;           // byte offset from gsrc
        asm volatile("global_load_async_to_lds_b128 %0, %1, %2"
                     :: "v"(lds), "v"(goff), "s"(gsrc) : "memory");
    }
    // W_hh pair-interleave: sW2[(k*16+j)*2 + p] = Whh[2j+p][k]
    if (tid < NG * HH) {
        const int k = tid >> 5, m = tid & 31;
        sW2[tid] = wsrc[m * HH + k];
    }
    asm volatile("s_wait_asynccnt 0" ::: "memory");
    __syncthreads();

    const int len = counts[widx];
    int ch[CC];
    {
        const int4* cp = (const int4*)(chars + widx * CC);
        int4 a = cp[0], b = cp[1], c2 = cp[2], d2 = cp[3];
        ch[0]=a.x;  ch[1]=a.y;  ch[2]=a.z;  ch[3]=a.w;
        ch[4]=b.x;  ch[5]=b.y;  ch[6]=b.z;  ch[7]=b.w;
        ch[8]=c2.x; ch[9]=c2.y; ch[10]=c2.z; ch[11]=c2.w;
        ch[12]=d2.x; ch[13]=d2.y; ch[14]=d2.z; ch[15]=d2.w;
    }

    float acc[HH];
    if (dir == 0) run_dir<false>(sG, sW2, ch, len, acc);   // block-uniform branch
    else          run_dir<true >(sG, sW2, ch, len, acc);

    float4* op = (float4*)(out + (size_t)widx * 16 + dir * HH);
    op[0] = make_float4(acc[0], acc[1], acc[2], acc[3]);
    op[1] = make_float4(acc[4], acc[5], acc[6], acc[7]);
}

extern "C" void kernel_launch(void* const* d_in, const int* in_sizes, int n_in,
                              void* d_out, int out_size, void* d_ws, size_t ws_size,
                              hipStream_t stream) {
    const int*   chars  = (const int*)d_in[0];
    const int*   counts = (const int*)d_in[1];
    const float* emb    = (const float*)d_in[2];
    const float* Wihf   = (const float*)d_in[3];
    const float* Whhf   = (const float*)d_in[4];
    const float* bihf   = (const float*)d_in[5];
    const float* bhhf   = (const float*)d_in[6];
    const float* Wihb   = (const float*)d_in[7];
    const float* Whhb   = (const float*)d_in[8];
    const float* bihb   = (const float*)d_in[9];
    const float* bhhb   = (const float*)d_in[10];

    float* G = (float*)d_ws;   // [2][272][32] f32 = 69,632 B

    build_tables_kernel<<<MT * 2 * 2, 32, 0, stream>>>(
        emb, Wihf, bihf, bhhf, Wihb, bihb, bhhb, G);

    encode_kernel<<<256, 256, 0, stream>>>(
        chars, counts, G, Whhf, Whhb, (float*)d_out);
}